// CTCInferenceLayer_79783312490906
// MI455X (gfx1250) — compile-verified
//
#include <hip/hip_runtime.h>

// CTC prefix beam search (T=256, B=32, C=64, beam 16, top 4), one wave32 per
// batch sample. Latency-bound 256-step dependent chain -> the entire step is
// register/shuffle-resident and barrier-free; LDS is used only as the target
// of the gfx1250 async DMA (global_load_async_to_lds_b64 + s_wait_asynccnt)
// double-buffering the next frame's logits, and for the write-only
// backpointer table used by the final traceback.

#define TMAX  256
#define CNUM  64
#define NBEAM 16
#define TOPK  4
#define NEGV  (-1.0e9f)
#define MASKV (-3.0e38f)

__device__ __forceinline__ float lae(float a, float b) {
  float m = fmaxf(a, b);
  return m + __logf(__expf(a - m) + __expf(b - m));
}

__global__ __launch_bounds__(32)
void ctc_beam_search_kernel(const float* __restrict__ data,
                            const int*   __restrict__ dlen,
                            float*       __restrict__ out,
                            int Bn)
{
  __shared__ __align__(16) float raw[2][CNUM];   // async DMA double buffer
  __shared__ int bp[TMAX][NBEAM];                // packed (parent<<8)|(char+1), 0=stay
  __shared__ int tmpseq[TMAX];

  const int lane = threadIdx.x;
  const int b    = blockIdx.x;
  const int j    = lane & 15;                    // beam id (lanes 16..31 mirror)

  int tlen = dlen[b];
  tlen = tlen < 0 ? 0 : (tlen > TMAX ? TMAX : tlen);

  // register-resident beam state: beam j lives in lane j (mirrored in j+16)
  int      ln = 0, lastj = -1, actj = (j == 0) ? 1 : 0;
  float    pbj = (j == 0) ? 0.0f : NEGV;
  float    pnbj = NEGV;
  unsigned h1j = 0u, h2j = 0u;

  const unsigned long long gbase =
      (unsigned long long)(const void*)(data + (size_t)b * CNUM);
  const unsigned fstride = (unsigned)(Bn * CNUM * 4);    // bytes per frame

  // prime: async DMA frame 0 into raw[0] (8 B/lane = 256 B)
  {
    unsigned la = (unsigned)(size_t)(const void*)&raw[0][0] + (unsigned)(lane * 8);
    unsigned vo = (unsigned)(lane * 8);
    asm volatile("global_load_async_to_lds_b64 %0, %1, %2"
                 :: "v"(la), "v"(vo), "s"(gbase) : "memory");
  }

  for (int t = 0; t < tlen; ++t) {
    const int cur = t & 1;

    // frame t ready once ASYNCcnt drains (async loads complete in order)
    asm volatile("s_wait_asynccnt 0" ::: "memory");
    float x0 = raw[cur][lane];
    float x1 = raw[cur][lane + 32];

    // kick DMA of frame t+1 into the other buffer; hidden behind this step
    if (t + 1 < tlen) {
      unsigned la = (unsigned)(size_t)(const void*)&raw[cur ^ 1][0] + (unsigned)(lane * 8);
      unsigned vo = (unsigned)(t + 1) * fstride + (unsigned)(lane * 8);
      asm volatile("global_load_async_to_lds_b64 %0, %1, %2"
                   :: "v"(la), "v"(vo), "s"(gbase) : "memory");
    }

    // ---- log-softmax over C=64, two values per lane, wave reductions ----
    float m = fmaxf(x0, x1);
    for (int off = 16; off; off >>= 1) m = fmaxf(m, __shfl_xor(m, off, 32));
    float s = __expf(x0 - m) + __expf(x1 - m);
    for (int off = 16; off; off >>= 1) s += __shfl_xor(s, off, 32);
    float ls  = __logf(s);
    float lp0 = x0 - m - ls;                 // lp[lane]
    float lp1 = x1 - m - ls;                 // lp[lane+32]

    // ---- stay candidates (beam j in lane j) ----
    float totj = lae(pbj, pnbj);
    float lpb  = __shfl(lp0, 0, 32);         // lp[BLANK]
    int   li   = lastj < 0 ? 0 : lastj;
    float l0   = __shfl(lp0, li & 31, 32);   // dynamic-lane gather of lp[last]
    float l1   = __shfl(lp1, li & 31, 32);
    float lpl  = (li >= 32) ? l1 : l0;
    float staypb  = totj + lpb;
    float staypnb = (ln > 0) ? (pnbj + lpl) : NEGV;

    // beam hash table broadcast into every lane's registers
    unsigned H1[NBEAM], H2[NBEAM];
    int      ACT[NBEAM];
#pragma unroll
    for (int q = 0; q < NBEAM; ++q) {
      H1[q]  = __shfl(h1j, q, 32);
      H2[q]  = __shfl(h2j, q, 32);
      ACT[q] = __shfl(actj, q, 32);
    }

    // ---- extension candidates: slot i of this lane = (b=i>>1, c=lane+32*(i&1)) ----
    float cand[32];
    float sums[NBEAM];                        // linear-domain merge accumulators
#pragma unroll
    for (int q = 0; q < NBEAM; ++q) sums[q] = 0.0f;

#pragma unroll
    for (int i = 0; i < 32; ++i) {
      const int bi = i >> 1;                  // compile-time constant
      const int c  = lane + ((i & 1) << 5);
      float lpc  = (i & 1) ? lp1 : lp0;
      float tb   = __shfl(totj,  bi, 32);
      float pb_b = __shfl(pbj,   bi, 32);
      int   lb   = __shfl(lastj, bi, 32);

      float extv = ((c == lb) ? pb_b : tb) + lpc;
      if (c == 0 || !ACT[bi]) extv = NEGV;    // no blank-extend, dead parents

      unsigned cc = (unsigned)(c + 1);
      unsigned c1 = H1[bi] * 1000003u + cc;
      unsigned c2 = H2[bi] * 69069u  + cc;

      float e = __expf(extv);                 // exp(NEGV)==0
      int any = 0;
#pragma unroll
      for (int q = 0; q < NBEAM; ++q) {
        int mt = (((c1 ^ H1[q]) | (c2 ^ H2[q])) == 0u) & (ACT[q] != 0);
        sums[q] += mt ? e : 0.0f;
        any |= mt;
      }
      cand[i] = any ? NEGV : extv;            // killed if merged into a beam
    }

    // ---- fold merged mass into stay_pnb of beam j ----
#pragma unroll
    for (int q = 0; q < NBEAM; ++q) {
      float sq = sums[q];
      for (int off = 16; off; off >>= 1) sq += __shfl_xor(sq, off, 32);
      if (j == q) staypnb = lae(staypnb, __logf(sq));   // log(0)=-inf -> no-op
    }
    float staytot = lae(staypb, staypnb);

    // stay slot: global index g = lane for lanes<16 (reference ordering),
    // mirrors carry an inert pad
    float cst = (lane < NBEAM) ? staytot : MASKV;
    int   gst = (lane < NBEAM) ? lane : (1040 + lane);

    // ---- top-16 of 1040, all in registers; used-mask replaces LDS masking ----
    unsigned long long used = 0ull;
    float selv = MASKV;
    int   selg = 0;
#pragma unroll 1
    for (int r = 0; r < NBEAM; ++r) {
      float tv[33];
      int   tg[33];
#pragma unroll
      for (int q = 0; q < 32; ++q) {
        tv[q] = (used & (1ull << q)) ? MASKV : cand[q];
        tg[q] = NBEAM + lane + (q << 5);      // global score index of ext slot
      }
      tv[32] = (used & (1ull << 32)) ? MASKV : cst;
      tg[32] = gst;
      // fold stay slot, then depth-5 tree over the 32 ext slots
      if (tv[32] > tv[0] || (tv[32] == tv[0] && tg[32] < tg[0])) { tv[0] = tv[32]; tg[0] = tg[32]; }
#pragma unroll
      for (int st = 16; st >= 1; st >>= 1)
#pragma unroll
        for (int k = 0; k < st; ++k)
          if (tv[k + st] > tv[k] || (tv[k + st] == tv[k] && tg[k + st] < tg[k])) {
            tv[k] = tv[k + st]; tg[k] = tg[k + st];
          }
      float v = tv[0];
      int   g = tg[0];
      for (int off = 16; off; off >>= 1) {
        float ov = __shfl_xor(v, off, 32);
        int   og = __shfl_xor(g, off, 32);
        if (ov > v || (ov == v && og < g)) { v = ov; g = og; }
      }
      if (j == r) { selv = v; selg = g; }     // lane r (and mirror) keep pick r
      if (g < NBEAM) {                        // mask winner in its owner lane
        if (j == g) used |= (1ull << 32);
      } else {
        int e = g - NBEAM;
        if (lane == (e & 31)) used |= (1ull << (e >> 5));
      }
    }

    // ---- rebuild beam state (pure shuffles from old registers) ----
    int iss = (selg < NBEAM) ? 1 : 0;
    int par = iss ? selg : ((selg - NBEAM) >> 6);
    int cn  = iss ? 0    : ((selg - NBEAM) & 63);
    int      pl    = __shfl(ln,      par, 32);
    float    ppb   = __shfl(staypb,  par, 32);
    float    ppnb  = __shfl(staypnb, par, 32);
    unsigned ph1   = __shfl(h1j,     par, 32);
    unsigned ph2   = __shfl(h2j,     par, 32);
    int      plast = __shfl(lastj,   par, 32);
    ln    = pl + (iss ? 0 : 1);
    pbj   = iss ? ppb  : NEGV;
    pnbj  = iss ? ppnb : selv;
    h1j   = iss ? ph1 : (ph1 * 1000003u + (unsigned)(cn + 1));
    h2j   = iss ? ph2 : (ph2 * 69069u  + (unsigned)(cn + 1));
    lastj = iss ? plast : cn;
    actj  = (selv > NEGV * 0.5f) ? 1 : 0;

    if (lane < NBEAM) bp[t][lane] = (par << 8) | (iss ? 0 : (cn + 1));  // write-only
  }

  // ---- final: top-4 beams, backpointer traceback ----
  __syncthreads();                            // bp[] now visible to lane 0
  float totf = (lane < NBEAM) ? lae(pbj, pnbj) : MASKV;
  float w = totf;
#pragma unroll 1
  for (int r = 0; r < TOPK; ++r) {
    float v = w;
    int   i = lane;
    for (int off = 16; off; off >>= 1) {
      float ov = __shfl_xor(v, off, 32);
      int   oi = __shfl_xor(i, off, 32);
      if (ov > v || (ov == v && oi < i)) { v = ov; i = oi; }
    }
    int leni = __shfl(ln, i, 32);
    if (lane == 0) {
      out[(size_t)b * TOPK + r]                     = -v;             // neg log prob
      out[(size_t)Bn * TOPK + (size_t)b * TOPK + r] = (float)leni;    // length
    }
    for (int p = lane; p < TMAX; p += 32) tmpseq[p] = -1;
    __syncthreads();
    if (lane == 0) {
      int pos = leni > TMAX ? TMAX : leni;
      int cu  = i;
      for (int tt = tlen - 1; tt >= 0; --tt) {
        int e  = bp[tt][cu];
        int ch = e & 255;
        if (ch) { --pos; if (pos >= 0) tmpseq[pos] = ch - 1; }
        cu = (e >> 8) & 255;
      }
    }
    __syncthreads();
    float* lab = out + (size_t)2 * Bn * TOPK + ((size_t)(b * TOPK + r)) * TMAX;
    for (int p = lane; p < TMAX; p += 32) lab[p] = (float)tmpseq[p];
    if (lane == i) w = MASKV;
    __syncthreads();
  }
}

extern "C" void kernel_launch(void* const* d_in, const int* in_sizes, int n_in,
                              void* d_out, int out_size, void* d_ws, size_t ws_size,
                              hipStream_t stream) {
  (void)n_in; (void)out_size; (void)d_ws; (void)ws_size;
  const float* data = (const float*)d_in[0];
  const int*   dlen = (const int*)d_in[1];
  float*       out  = (float*)d_out;
  const int Bn = in_sizes[1];                 // batch = #blocks, one wave32 each
  ctc_beam_search_kernel<<<dim3(Bn), dim3(32), 0, stream>>>(data, dlen, out, Bn);
}